// GPTCompress_65034394796617
// MI455X (gfx1250) — compile-verified
//
#include <hip/hip_runtime.h>
#include <math.h>

typedef __attribute__((ext_vector_type(2))) float v2f;
typedef __attribute__((ext_vector_type(8))) float v8f;

#define BATCH   8
#define LEN_IN  4096
#define DMODEL  128
#define LEN_OUT 2048
#define KC      128      // K columns per chunk
#define LDSTR   132      // padded LDS row stride (floats): 528B rows, 16B aligned
#define TILE_M  16

// One workgroup (256 threads = 8 wave32) per (batch, 16-output-row tile).
// Phase 1: band-limited row denominators. Phase 2: per-128-column chunk,
// stage normalized weights in LDS (b128), store them to global (b128), and
// (band chunks only) accumulate z' = W x via V_WMMA_F32_16X16X4_F32.
__global__ __launch_bounds__(256)
void gpt_compress_fused(const float* __restrict__ x,      // [B, LEN_IN, D]
                        const float* __restrict__ deltas, // [B]
                        const float* __restrict__ sigma,  // [1]
                        float* __restrict__ zout,         // [B, LEN_OUT, D]
                        float* __restrict__ pmout,        // [B, LEN_OUT]
                        float* __restrict__ wout)         // [B, LEN_OUT, LEN_IN]
{
    __shared__ float Wlds[TILE_M * LDSTR];
    __shared__ float red[TILE_M][16];
    __shared__ float denomS[TILE_M];

    const int b    = blockIdx.y;
    const int o0   = blockIdx.x * TILE_M;
    const int tid  = threadIdx.x;
    const int lane = tid & 31;
    const int wave = tid >> 5;

    // lens from deltas (deltas = LEN_OUT - lens, exact small integers in f32)
    const float delta   = deltas[b];
    const float lens_f  = (float)LEN_OUT - delta;
    const int   lens_i  = (int)(lens_f + 0.5f);
    const float changed = lens_f + delta;             // == LEN_OUT
    const float scale   = lens_f / changed;
    const float es      = __expf(sigma[0]);
    const float inv2s   = 0.5f / es;
    // |d| beyond margin -> exp underflows to exact f32 zero (incl. denormals)
    const float margin  = sqrtf(210.0f * es) + 2.0f;

    const float muLo = (float)o0 * scale;
    const float muHi = (float)(o0 + TILE_M - 1) * scale;
    int iLo = (int)floorf(muLo - margin); if (iLo < 0) iLo = 0;
    int iHi = (int)ceilf (muHi + margin) + 1; if (iHi > LEN_IN) iHi = LEN_IN;

    // ---------------- pass 1: denominators (band-limited) ----------------
    {
        const int r   = tid >> 4;        // row 0..15
        const int l16 = tid & 15;        // 16 threads per row
        const int o   = o0 + r;
        const float mu = (float)o * scale;
        const bool opad = ((float)o >= changed);
        float s = 0.0f;
        if (!opad) {
            for (int i = iLo + l16; i < iHi; i += 16) {
                if (i < lens_i) {
                    float d = (float)i - mu;
                    s += __expf(-d * d * inv2s);
                }
            }
        }
        red[r][l16] = s;
    }
    __syncthreads();
    if (tid < TILE_M) {
        float s = 0.0f;
        #pragma unroll
        for (int j = 0; j < 16; ++j) s += red[tid][j];
        denomS[tid] = (s == 0.0f) ? 1.0f : s;
        const int o = o0 + tid;
        pmout[b * LEN_OUT + o] = ((float)o >= changed) ? 1.0f : 0.0f;
    }
    __syncthreads();

    // ---------------- pass 2: weight stores + WMMA ----------------
    const float* xb = x    + (size_t)b * LEN_IN * DMODEL;
    float*       wb = wout + (size_t)(b * LEN_OUT + o0) * LEN_IN;
    const int n0 = wave * 16;            // each wave owns 16 columns of d
    v8f c = {};                          // 16x16 f32 accumulator tile

    const float4 zero4 = make_float4(0.0f, 0.0f, 0.0f, 0.0f);

    for (int k0 = 0; k0 < LEN_IN; k0 += KC) {
        const bool contrib = (k0 < iHi) && (k0 + KC > iLo);  // block-uniform
        if (contrib) {
            // fill LDS with normalized weights; mirror to global (16B vectors)
            #pragma unroll
            for (int it = 0; it < 2; ++it) {
                const int q   = tid + it * 256;      // float4 index, 512 total
                const int r   = q >> 5;              // 32 float4 per row
                const int kk4 = (q & 31) * 4;
                const int o   = o0 + r;
                const float mu  = (float)o * scale;
                const float inv = 1.0f / denomS[r];
                const bool  orow = ((float)o < changed);
                float wv[4];
                #pragma unroll
                for (int e = 0; e < 4; ++e) {
                    const int i = k0 + kk4 + e;
                    float w = 0.0f;
                    if (orow && (i < lens_i)) {
                        float d = (float)i - mu;
                        w = __expf(-d * d * inv2s) * inv;
                    }
                    wv[e] = w;
                }
                const float4 w4 = make_float4(wv[0], wv[1], wv[2], wv[3]);
                *reinterpret_cast<float4*>(&Wlds[r * LDSTR + kk4]) = w4;
                *reinterpret_cast<float4*>(&wb[(size_t)r * LEN_IN + k0 + kk4]) = w4;
            }
            __syncthreads();
            // A-frag: lane holds row M=lane&15, K pair {0,1} (lanes 0-15)
            // or {2,3} (lanes 16-31). B-frag mirrors K split, N = lane&15.
            const int m    = lane & 15;
            const int koff = (lane >> 4) * 2;
            for (int ks = 0; ks < KC / 4; ++ks) {
                const int kb = ks * 4 + koff;
                v2f a, bf;
                a.x = Wlds[m * LDSTR + kb];
                a.y = Wlds[m * LDSTR + kb + 1];
                const int gk = k0 + kb;
                bf.x = xb[(size_t)gk       * DMODEL + n0 + m];
                bf.y = xb[(size_t)(gk + 1) * DMODEL + n0 + m];
                c = __builtin_amdgcn_wmma_f32_16x16x4_f32(
                        false, a, false, bf, (short)0, c, false, false);
            }
            __syncthreads();   // protect Wlds before next chunk's fill
        } else {
            // outside the Gaussian band: weights are exact zeros; wide stores
            #pragma unroll
            for (int it = 0; it < 2; ++it) {
                const int q   = tid + it * 256;
                const int r   = q >> 5;
                const int kk4 = (q & 31) * 4;
                *reinterpret_cast<float4*>(&wb[(size_t)r * LEN_IN + k0 + kk4]) = zero4;
            }
        }
    }

    // ---------------- write z' from C/D fragment layout ----------------
    {
        const int mbase = (lane >> 4) * 8;   // lanes 16-31 hold M = vr + 8
        const int n     = n0 + (lane & 15);
        float* zb = zout + (size_t)(b * LEN_OUT + o0) * DMODEL;
        #pragma unroll
        for (int vr = 0; vr < 8; ++vr) {
            zb[(size_t)(mbase + vr) * DMODEL + n] = c[vr];
        }
    }
}

extern "C" void kernel_launch(void* const* d_in, const int* in_sizes, int n_in,
                              void* d_out, int out_size, void* d_ws, size_t ws_size,
                              hipStream_t stream) {
    (void)in_sizes; (void)n_in; (void)d_ws; (void)ws_size; (void)out_size;
    const float* x      = (const float*)d_in[0];
    // d_in[1] is x_mask (bool); semantics reconstructed exactly from deltas.
    const float* deltas = (const float*)d_in[2];
    const float* sigma  = (const float*)d_in[3];
    // d_in[4] is len_out scalar (2048), fixed by problem shape.

    float* zout  = (float*)d_out;                                   // [8,2048,128]
    float* pmout = zout + (size_t)BATCH * LEN_OUT * DMODEL;         // [8,2048]
    float* wout  = pmout + (size_t)BATCH * LEN_OUT;                 // [8,2048,4096]

    dim3 grid(LEN_OUT / TILE_M, BATCH);   // (128, 8)
    dim3 block(256);                      // 8 wave32
    gpt_compress_fused<<<grid, block, 0, stream>>>(x, deltas, sigma,
                                                   zout, pmout, wout);
}